// ARXCell_23862838296939
// MI455X (gfx1250) — compile-verified
//
#include <hip/hip_runtime.h>

// ARX cell: a2 = iw·itdl + ow·otdl + b ; shift-register updates of itdl/otdl.
// Memory-bound: ~280 MB of traffic, ~69 MFLOP. Stream itdl once; fuse the
// shifted copy with the dot. Dot runs on the fp32 WMMA unit (16x16x4).

typedef __attribute__((ext_vector_type(2))) float v2f;
typedef __attribute__((ext_vector_type(4))) float v4f;
typedef __attribute__((ext_vector_type(8))) float v8f;

#define BATCH   16384
#define DI      2048      // input delay line length
#define DO      64        // output delay line length
#define IN_SZ   64        // new input samples appended per step

#define WAVES   4
#define TPB     (WAVES * 32)
#define ROWS    16                 // batches per block (WMMA M)
#define KPW     (DI / WAVES)       // 512 K per wave
#define TILE_K  128
#define TILES   (KPW / TILE_K)     // 4
#define LDS_STRIDE (TILE_K + 4)    // 132 floats: pad -> conflict-free ds_load_b64

__global__ __launch_bounds__(TPB) void arx_cell_kernel(
    const float* __restrict__ input,   // [B,64]
    const float* __restrict__ itdl,    // [B,2048]
    const float* __restrict__ otdl,    // [B,64]
    const float* __restrict__ iw,      // [2048]
    const float* __restrict__ ow,      // [64]
    const float* __restrict__ bias,    // [1]
    float* __restrict__ out_a2,        // [B]
    float* __restrict__ out_itdl,      // [B,2048]
    float* __restrict__ out_otdl)      // [B,64]
{
    __shared__ float iw_lds[DI];                         // 8 KB
    __shared__ float tile[WAVES][ROWS][LDS_STRIDE];      // 33 KB
    __shared__ float red[WAVES][32][8];                  // 4 KB
    __shared__ float dot_lds[ROWS];

    const int tid  = threadIdx.x;
    const int lane = tid & 31;
    const int wave = tid >> 5;
    const int b0   = blockIdx.x * ROWS;

    // Stage iw into LDS (block-wide), coalesced b128.
    #pragma unroll
    for (int j = 0; j < DI / (TPB * 4); ++j) {
        const int idx = (j * TPB + tid) * 4;
        *(v4f*)&iw_lds[idx] = *(const v4f*)&iw[idx];
    }
    __syncthreads();

    const int g   = lane >> 4;   // WMMA lane group (K offset +2 for lanes 16-31)
    const int r16 = lane & 15;   // row within A tile

    v8f c = {0.f, 0.f, 0.f, 0.f, 0.f, 0.f, 0.f, 0.f};

    const int kbase = wave * KPW;
    for (int t = 0; t < TILES; ++t) {
        const int k0 = kbase + t * TILE_K;
        const int kc = k0 + 4 * lane;          // this lane's K column (b128)

        // Phase 1: coalesced stream of a 16x128 fp32 tile.
        // Each element is read ONCE from HBM: registers feed both the shifted
        // store to itdl_new (k-64) and the LDS stage for the WMMA A operand.
        #pragma unroll
        for (int i = 0; i < ROWS; ++i) {
            const size_t row = (size_t)(b0 + i) * DI;
            v4f gv = *(const v4f*)&itdl[row + kc];
            if (kc >= IN_SZ)                    // taps with k<64 are dropped
                *(v4f*)&out_itdl[row + kc - IN_SZ] = gv;
            *(v4f*)&tile[wave][i][4 * lane] = gv;
        }

        // Phase 2: 32x V_WMMA_F32_16X16X4_F32, accumulating the K slice.
        // A: 16 batches x 4 K from LDS (ISA 16x4 fp32 layout: lanes0-15 K=0/1,
        //    lanes16-31 K=2/3). B: iw[k] broadcast to all 16 columns, so every
        //    D column carries the same per-batch partial dot.
        #pragma unroll
        for (int kk = 0; kk < TILE_K; kk += 4) {
            v2f a  = *(const v2f*)&tile[wave][r16][kk + 2 * g];
            v2f bv = *(const v2f*)&iw_lds[k0 + kk + 2 * g];
            c = __builtin_amdgcn_wmma_f32_16x16x4_f32(
                    false, a, false, bv, (short)0, c, false, false);
        }
    }

    // Cross-wave reduction of the K-split partials (D layouts are lane-identical).
    #pragma unroll
    for (int r = 0; r < 8; ++r) red[wave][lane][r] = c[r];
    __syncthreads();

    if (wave == 0) {
        #pragma unroll
        for (int r = 0; r < 8; ++r)
            c[r] = red[0][lane][r] + red[1][lane][r] +
                   red[2][lane][r] + red[3][lane][r];

        // D[m,n]: vgpr r, lanes0-15 -> m=r, lanes16-31 -> m=r+8 (all n equal).
        if (r16 == 0) {
            #pragma unroll
            for (int r = 0; r < 8; ++r) dot_lds[r + 8 * g] = c[r];
        }

        const float owx = ow[2 * lane];
        const float owy = ow[2 * lane + 1];
        const float bb  = bias[0];

        for (int i = 0; i < ROWS; ++i) {
            const int b = b0 + i;
            // otdl row: 64 floats, 2 per lane (coalesced b64).
            v2f ov = *(const v2f*)&otdl[(size_t)b * DO + 2 * lane];
            float p = ov.x * owx + ov.y * owy;
            #pragma unroll
            for (int off = 16; off > 0; off >>= 1)
                p += __shfl_xor(p, off, 32);

            const float a2 = dot_lds[i] + p + bb;

            // otdl shift by 1: element j goes to j-1; slot 63 gets a2.
            if (lane > 0) out_otdl[(size_t)b * DO + 2 * lane - 1] = ov.x;
            out_otdl[(size_t)b * DO + 2 * lane] = ov.y;
            if (lane == 0) {
                out_otdl[(size_t)b * DO + 63] = a2;
                out_a2[b] = a2;
            }

            // itdl tail: append the 64 new input samples.
            v2f iv = *(const v2f*)&input[(size_t)b * IN_SZ + 2 * lane];
            *(v2f*)&out_itdl[(size_t)b * DI + (DI - IN_SZ) + 2 * lane] = iv;
        }
    }
}

extern "C" void kernel_launch(void* const* d_in, const int* in_sizes, int n_in,
                              void* d_out, int out_size, void* d_ws, size_t ws_size,
                              hipStream_t stream) {
    (void)in_sizes; (void)n_in; (void)d_ws; (void)ws_size; (void)out_size;

    const float* input = (const float*)d_in[0];
    const float* itdl  = (const float*)d_in[1];
    const float* otdl  = (const float*)d_in[2];
    const float* iw    = (const float*)d_in[3];
    const float* ow    = (const float*)d_in[4];
    const float* bias  = (const float*)d_in[5];

    float* out      = (float*)d_out;
    float* out_a2   = out;                               // [B]
    float* out_itdl = out + BATCH;                       // [B*DI]
    float* out_otdl = out + BATCH + (size_t)BATCH * DI;  // [B*DO]

    dim3 grid(BATCH / ROWS);   // 1024 blocks
    dim3 block(TPB);           // 128 threads = 4 waves
    arx_cell_kernel<<<grid, block, 0, stream>>>(
        input, itdl, otdl, iw, ow, bias, out_a2, out_itdl, out_otdl);
}